// NUFFTOp_50465865728375
// MI455X (gfx1250) — compile-verified
//
#include <hip/hip_runtime.h>
#include <hip/hip_bf16.h>
#include <math.h>

// ---------------------------------------------------------------------------
// NUFFT forward (type-2) for MI455X / gfx1250.
//   kgrid = A (512x256 complex) * img (256x256 real) * A^T   via f32 WMMA
//   then 6x6 Kaiser-Bessel gather at 1M non-uniform points (L2-resident grid).
// A folds de-apodization + centered-pad + ifftshift:  A[k,n] = a(n)*e^{-2pi i k(n-128)/512}
// ---------------------------------------------------------------------------

typedef float v2f __attribute__((ext_vector_type(2)));
typedef float v8f __attribute__((ext_vector_type(8)));

#define GSZ 512      // oversampled grid
#define NSZ 256      // image size
#define NB  8        // batches
#define NM  131072   // points per batch
#define KB_BETA 13.8551015f   // pi*sqrt((6/2)^2*(2-0.5)^2 - 0.8)

// ---------------- Kaiser-Bessel helpers (A&S 9.8.1 / 9.8.2) ----------------
__device__ __forceinline__ float i0f_dev(float z) {
    if (z < 3.75f) {
        float t = z * (1.0f / 3.75f);
        float t2 = t * t;
        return 1.0f + t2*(3.5156229f + t2*(3.0899424f + t2*(1.2067492f +
                     t2*(0.2659732f + t2*(0.0360768f + t2*0.0045813f)))));
    } else {
        float it = 3.75f / z;
        float p = 0.39894228f + it*(0.01328592f + it*(0.00225319f + it*(-0.00157565f +
                  it*(0.00916281f + it*(-0.02057706f + it*(0.02635537f + it*(-0.01647633f +
                  it*0.00392377f)))))));
        return __expf(z) * __frsqrt_rn(z) * p;
    }
}

__device__ __forceinline__ float kb_w(float u) {
    float x = u * (1.0f / 3.0f);      // 2u/J, J=6
    float t = 1.0f - x * x;
    if (t <= 0.0f) return 0.0f;
    return i0f_dev(KB_BETA * sqrtf(t));
}

// ---------------- 1) DFT matrix with folded apodization --------------------
__global__ void __launch_bounds__(256)
build_dft(float* __restrict__ Ar, float* __restrict__ Ai) {
    int id = blockIdx.x * 256 + threadIdx.x;     // 512*256 elements
    int k = id >> 8;
    int n = id & 255;
    // de-apodization coefficient a(n) = sqrt(t)/sinh(sqrt(t)), t = beta^2 - arg^2
    float x   = (float)n - 128.0f;
    float arg = 3.14159265358979f * 6.0f * x * (1.0f / 512.0f);
    float t   = KB_BETA * KB_BETA - arg * arg;   // > 0 for this config
    float s   = sqrtf(t);
    float a   = s / sinhf(s);
    // phase: exact integer reduction mod 512 (exp is periodic)
    int p = (k * (n - 128)) & 511;
    float th = -(float)p * (6.283185307179586f / 512.0f);
    float sn, cs;
    __sincosf(th, &sn, &cs);
    Ar[id] = a * cs;
    Ai[id] = a * sn;
}

// ---------------- 2) Stage 1: T = A * img  (f32 WMMA) ----------------------
// One wave32 per 16x16 tile of T (512x256 per batch). K = 256 in steps of 4.
__global__ void __launch_bounds__(256)
stage1(const float* __restrict__ img, const float* __restrict__ Ar,
       const float* __restrict__ Ai, float* __restrict__ Tr, float* __restrict__ Ti) {
    int wave = (blockIdx.x * 256 + threadIdx.x) >> 5;
    int lane = threadIdx.x & 31;
    int b  = wave >> 9;              // 512 tiles per batch
    int t  = wave & 511;
    int tm = t >> 4;                 // 0..31  (rows of 512)
    int tn = t & 15;                 // 0..15  (cols of 256)
    int grp = lane >> 4;
    int row = lane & 15;

    const float* imgb = img + b * (NSZ * NSZ);
    int arow = (tm * 16 + row) * NSZ;   // A-fragment row (M = output row)
    int bcol = tn * 16 + row;           // B-fragment column (N = lane&15)

    v8f cr = {}; v8f ci = {};
    for (int k0 = 0; k0 < NSZ; k0 += 4) {
        int ka = k0 + 2 * grp;                      // VGPR0 -> K=2g, VGPR1 -> K=2g+1
        v2f fa_r, fa_i, fb;
        fa_r.x = Ar[arow + ka];     fa_r.y = Ar[arow + ka + 1];
        fa_i.x = Ai[arow + ka];     fa_i.y = Ai[arow + ka + 1];
        fb.x   = imgb[ka * NSZ + bcol];
        fb.y   = imgb[(ka + 1) * NSZ + bcol];
        cr = __builtin_amdgcn_wmma_f32_16x16x4_f32(false, fa_r, false, fb, (short)0, cr, false, false);
        ci = __builtin_amdgcn_wmma_f32_16x16x4_f32(false, fa_i, false, fb, (short)0, ci, false, false);
    }
    float* trb = Tr + b * (GSZ * NSZ);
    float* tib = Ti + b * (GSZ * NSZ);
    int c = tn * 16 + row;
    #pragma unroll
    for (int v = 0; v < 8; ++v) {
        int r = tm * 16 + v + 8 * grp;              // C/D layout: VGPR v -> M = v + 8*grp
        trb[r * NSZ + c] = cr[v];
        tib[r * NSZ + c] = ci[v];
    }
}

// ---------------- 3) Stage 2: kgrid = T * A^T  (f32 WMMA) ------------------
// Complex*complex: f32 WMMA has no A/B negate (NEG is CNeg-only), so keep
// Tr*Br and Ti*Bi in separate accumulators and subtract with VALU at the end.
__global__ void __launch_bounds__(256)
stage2(const float* __restrict__ Tr, const float* __restrict__ Ti,
       const float* __restrict__ Ar, const float* __restrict__ Ai,
       float2* __restrict__ kg) {
    int wave = (blockIdx.x * 256 + threadIdx.x) >> 5;
    int lane = threadIdx.x & 31;
    int b  = wave >> 10;             // 1024 tiles per batch
    int t  = wave & 1023;
    int tm = t >> 5;                 // 0..31
    int tn = t & 31;                 // 0..31
    int grp = lane >> 4;
    int row = lane & 15;

    const float* trb = Tr + b * (GSZ * NSZ);
    const float* tib = Ti + b * (GSZ * NSZ);
    int arow  = (tm * 16 + row) * NSZ;      // T row for A-fragment
    int bkrow = (tn * 16 + row) * NSZ;      // B[K=n2, N=k2] = A[k2*256 + n2]

    v8f crr = {}; v8f cii = {}; v8f cim = {};
    for (int k0 = 0; k0 < NSZ; k0 += 4) {
        int ka = k0 + 2 * grp;
        v2f a_r, a_i, b_r, b_i;
        a_r.x = trb[arow + ka];   a_r.y = trb[arow + ka + 1];
        a_i.x = tib[arow + ka];   a_i.y = tib[arow + ka + 1];
        b_r.x = Ar[bkrow + ka];   b_r.y = Ar[bkrow + ka + 1];
        b_i.x = Ai[bkrow + ka];   b_i.y = Ai[bkrow + ka + 1];
        crr = __builtin_amdgcn_wmma_f32_16x16x4_f32(false, a_r, false, b_r, (short)0, crr, false, false);
        cii = __builtin_amdgcn_wmma_f32_16x16x4_f32(false, a_i, false, b_i, (short)0, cii, false, false);
        cim = __builtin_amdgcn_wmma_f32_16x16x4_f32(false, a_r, false, b_i, (short)0, cim, false, false);
        cim = __builtin_amdgcn_wmma_f32_16x16x4_f32(false, a_i, false, b_r, (short)0, cim, false, false);
    }
    float2* kgb = kg + b * (GSZ * GSZ);
    int c = tn * 16 + row;
    #pragma unroll
    for (int v = 0; v < 8; ++v) {
        int r = tm * 16 + v + 8 * grp;
        float2 val;
        val.x = crr[v] - cii[v];
        val.y = cim[v];
        kgb[r * GSZ + c] = val;     // interleaved complex -> 8B gather later
    }
}

// ---------------- 4) KB gather interpolation -------------------------------
__global__ void __launch_bounds__(256)
interp(const float2* __restrict__ kg, const float* __restrict__ ktraj,
       const float* __restrict__ dcf, float* __restrict__ out) {
    int gid = blockIdx.x * 256 + threadIdx.x;    // NB*NM threads
    int b = gid >> 17;
    int m = gid & (NM - 1);

    const float sc = 512.0f / 6.283185307179586f;
    float t1 = ktraj[b * 2 * NM + m]      * sc;
    float t2 = ktraj[b * 2 * NM + NM + m] * sc;
    float f1 = floorf(t1), f2 = floorf(t2);
    int   b1 = (int)f1,    b2 = (int)f2;

    float w1[6], w2[6];
    int   i2a[6];
    #pragma unroll
    for (int j = 0; j < 6; ++j) {               // offsets -2..3
        float off = (float)(j - 2);
        w1[j]  = kb_w(t1 - (f1 + off));
        w2[j]  = kb_w(t2 - (f2 + off));
        i2a[j] = (b2 + j - 2) & (GSZ - 1);
    }

    const float2* kgb = kg + b * (GSZ * GSZ);
    float re = 0.0f, im = 0.0f;
    #pragma unroll
    for (int a = 0; a < 6; ++a) {
        int r = (b1 + a - 2) & (GSZ - 1);
        const float2* rowp = kgb + r * GSZ;
        float wa = w1[a];
        #pragma unroll
        for (int c2 = 0; c2 < 6; ++c2) {
            float2 v = rowp[i2a[c2]];           // L2-resident (kgrid = 16 MB < 192 MB L2)
            float w = wa * w2[c2];
            re += v.x * w;
            im += v.y * w;
        }
    }
    float d = dcf[b * NM + m];
    out[b * NM + m]           = re * d;         // plane 0: real
    out[NB * NM + b * NM + m] = im * d;         // plane 1: imag
}

// ---------------------------------------------------------------------------
extern "C" void kernel_launch(void* const* d_in, const int* in_sizes, int n_in,
                              void* d_out, int out_size, void* d_ws, size_t ws_size,
                              hipStream_t stream) {
    const float* image = (const float*)d_in[0];   // (8,256,256) f32
    const float* ktraj = (const float*)d_in[1];   // (8,2,131072) f32
    const float* dcf   = (const float*)d_in[2];   // (8,131072) f32
    float* out = (float*)d_out;                   // (2,8,131072) f32

    // workspace carve-up (floats): A 2x(512*256), T 2x(8*512*256), kgrid 8*512*512*2
    float* Ar = (float*)d_ws;
    float* Ai = Ar + GSZ * NSZ;
    float* Tr = Ai + GSZ * NSZ;
    float* Ti = Tr + NB * GSZ * NSZ;
    float2* kg = (float2*)(Ti + NB * GSZ * NSZ);  // 8-byte aligned (offset 9437184 B)

    build_dft<<<GSZ * NSZ / 256, 256, 0, stream>>>(Ar, Ai);
    stage1  <<<NB * 512 / 8,     256, 0, stream>>>(image, Ar, Ai, Tr, Ti);   // 4096 waves
    stage2  <<<NB * 1024 / 8,    256, 0, stream>>>(Tr, Ti, Ar, Ai, kg);      // 8192 waves
    interp  <<<NB * NM / 256,    256, 0, stream>>>(kg, ktraj, dcf, out);
}